// PointCloudExpansion_83837761618201
// MI455X (gfx1250) — compile-verified
//
#include <hip/hip_runtime.h>

typedef float v2f __attribute__((ext_vector_type(2)));
typedef float v8f __attribute__((ext_vector_type(8)));

namespace {
constexpr int kB = 8, kH = 16, kW = 16, kN = 32, kM = 64, kC = 4;
constexpr int kHW = kH * kW;
constexpr int kComb = kM + kN;      // 96 candidate points per neighbor
constexpr int kTiles = kComb / 16;  // 6 WMMA tiles of 16 rows
}

__global__ __launch_bounds__(256)
void pce_kernel(const float* __restrict__ selected,
                const float* __restrict__ weights,
                const float* __restrict__ original,
                float* __restrict__ ef /* B*HW*M*C, doubles as expanded_flat */) {
  __shared__ float4 comb[8][kComb];  // per-wave compacted candidates
  __shared__ float  nbuf[8][kComb];  // per-wave squared xyz norms
  __shared__ float4 newp[8][kM];     // per-wave staged new pillar

  const int b = (int)blockIdx.x;
  const int tid = (int)threadIdx.x;
  const int lane = tid & 31;
  const int wv = tid >> 5;  // wave id == neighbor id (0..7)

  float4* efb = (float4*)(ef + (size_t)b * kHW * kM * kC);
  const float4* orig_b = (const float4*)(original + (size_t)b * kHW * kM * kC);

  // Initialize working expanded_flat = original (also fully defines d_out).
  for (int i = tid; i < kHW * kM; i += 256) efb[i] = orig_b[i];
  __threadfence();
  __syncthreads();

  // Neighbor offset for this wave: 3x3 grid skipping center.
  const int k8 = (wv < 4) ? wv : wv + 1;
  const int dh = k8 / 3 - 1;
  const int dw = k8 % 3 - 1;

  const int halfsel = (lane >> 4) & 1;  // 0: lanes 0-15 (K0,K1 / M+0), 1: lanes 16-31 (K2,K3 / M+8)
  const int mlo = lane & 15;

  for (int p = 0; p < kHW; ++p) {
    const int h = p >> 4, x = p & 15;

    // Current pillar points/weights (one point per lane, N=32).
    float4 cp = ((const float4*)selected)[((size_t)b * kHW + p) * kN + lane];
    float cw = weights[((size_t)b * kHW + p) * kN + lane];
    const bool vval = cw > 0.0f;
    const unsigned vm = (unsigned)__ballot(vval);
    const int cnt_v = __popc(vm);

    if (cnt_v != 0) {  // block-uniform; !any_valid => step is a no-op
      const int nh = h + dh, nw = x + dw;
      const bool ok = (nh >= 0) && (nh < kH) && (nw >= 0) && (nw < kW);
      const int ch = nh < 0 ? 0 : (nh >= kH ? kH - 1 : nh);
      const int cx = nw < 0 ? 0 : (nw >= kW ? kW - 1 : nw);
      const int nidx = ch * kW + cx;

      // Zero this wave's candidate buffer (96 slots).
      comb[wv][lane]      = make_float4(0.f, 0.f, 0.f, 0.f);
      comb[wv][lane + 32] = make_float4(0.f, 0.f, 0.f, 0.f);
      comb[wv][lane + 64] = make_float4(0.f, 0.f, 0.f, 0.f);

      // Stable-compact the neighbor's 64 stored points (valid = channel-sum != 0).
      int cnt_n = 0;
      for (int j = 0; j < 2; ++j) {
        float4 pt = efb[(size_t)nidx * kM + j * 32 + lane];
        bool nv = (((pt.x + pt.y) + pt.z) + pt.w) != 0.0f;
        unsigned mm = (unsigned)__ballot(nv);
        int pos = cnt_n + __popc(mm & ((1u << lane) - 1u));
        if (nv) comb[wv][pos] = pt;
        cnt_n += __popc(mm);
      }
      // Stable-compact current pillar's valid points into slots [64, 64+cnt_v).
      {
        int pos = kM + __popc(vm & ((1u << lane) - 1u));
        if (vval) comb[wv][pos] = cp;
      }
      // Squared xyz norms (distance uses first 3 channels only).
      for (int j = 0; j < 3; ++j) {
        int i = j * 32 + lane;
        float4 q = comb[wv][i];
        nbuf[wv][i] = q.x * q.x + q.y * q.y + q.z * q.z;
      }

      const int count = cnt_n + cnt_v;
      if (count > kM) {
        // ---- FPS via WMMA: d = n_m + n_f - 2 * (x_m . f) ----
        // A operands: 6 tiles of 16x4 f32 (K3 = w-channel, killed by B's zero).
        v2f A[kTiles];
        float nrm[kTiles][8];
        float dist[kTiles][8];
#pragma unroll
        for (int t = 0; t < kTiles; ++t) {
          const float* pp = (const float*)&comb[wv][t * 16 + mlo];
          A[t] = *(const v2f*)(pp + halfsel * 2);  // (K0,K1) or (K2,K3)
#pragma unroll
          for (int r = 0; r < 8; ++r) {
            int rr = t * 16 + halfsel * 8 + r;
            nrm[t][r] = nbuf[wv][rr];
            bool val = (rr < kM) ? (rr < cnt_n) : (rr - kM < cnt_v);
            dist[t][r] = val ? 1.0e10f : -1.0f;  // matches reference dist0
          }
        }
        int far = (cnt_n > 0) ? 0 : kM;  // argmax(mask) = first valid index
        for (int s = 0; s < kM; ++s) {
          float4 f = comb[wv][far];
          if (lane == 0) newp[wv][s] = f;  // scan output = incoming `far`
          if (s == kM - 1) break;          // last update is discarded by scan

          float nf = f.x * f.x + f.y * f.y + f.z * f.z;
          v2f Bv;  // broadcast f to all 16 columns; K3 = 0 kills w-channel
          Bv.x = halfsel ? f.z : f.x;
          Bv.y = halfsel ? 0.0f : f.y;

          float bestv = -2.0f;  // below the -1 invalid floor
          int besti = 0;
#pragma unroll
          for (int t = 0; t < kTiles; ++t) {
            v8f acc = {};
            acc = __builtin_amdgcn_wmma_f32_16x16x4_f32(
                /*neg_a=*/false, A[t], /*neg_b=*/false, Bv,
                /*c_mod=*/(short)0, acc, /*reuse_a=*/false, /*reuse_b=*/false);
#pragma unroll
            for (int r = 0; r < 8; ++r) {
              float g = acc[r];
              float d = nrm[t][r] + nf - 2.0f * g;
              d = d < 0.0f ? 0.0f : d;       // true distance is >= 0
              float od = dist[t][r];
              float ndv = d < od ? d : od;   // invalid (-1) never updated since d>=0
              dist[t][r] = ndv;
              int rr = t * 16 + halfsel * 8 + r;
              if (ndv > bestv) { bestv = ndv; besti = rr; }  // first-index tie-break
            }
          }
          // Merge the two half-wave argmax candidates (rows interleave by 8).
          float ov = __shfl_xor(bestv, 16, 32);
          int oi = __shfl_xor(besti, 16, 32);
          if ((ov > bestv) || ((ov == bestv) && (oi < besti))) { bestv = ov; besti = oi; }
          far = besti;  // uniform across the wave
        }
      } else {
        // small path: first M entries of the compacted combined set.
        for (int j = 0; j < 2; ++j) {
          int i = j * 32 + lane;
          float4 o;
          if (i < cnt_n)                o = comb[wv][i];
          else if (i - cnt_n < cnt_v)   o = comb[wv][kM + (i - cnt_n)];
          else                          o = make_float4(0.f, 0.f, 0.f, 0.f);
          newp[wv][i] = o;
        }
      }

      // All reads of ef (incl. clipped-alias reads) complete before any write.
      __syncthreads();
      if (ok) {
        efb[(size_t)nidx * kM + lane]      = newp[wv][lane];
        efb[(size_t)nidx * kM + lane + 32] = newp[wv][lane + 32];
      }
      __threadfence();
      __syncthreads();
    }
  }
}

extern "C" void kernel_launch(void* const* d_in, const int* in_sizes, int n_in,
                              void* d_out, int out_size, void* d_ws, size_t ws_size,
                              hipStream_t stream) {
  (void)in_sizes; (void)n_in; (void)d_ws; (void)ws_size; (void)out_size;
  const float* selected = (const float*)d_in[0];
  const float* weights  = (const float*)d_in[1];
  const float* original = (const float*)d_in[2];
  float* out = (float*)d_out;
  pce_kernel<<<dim3(kB), dim3(256), 0, stream>>>(selected, weights, original, out);
}